// DeconvolutionEnergyScoreLoss_9337258901604
// MI455X (gfx1250) — compile-verified
//
#include <hip/hip_runtime.h>

typedef __attribute__((ext_vector_type(2))) float v2f;
typedef __attribute__((ext_vector_type(8))) float v8f;

#define N_ROWS    4096
#define G_GROUPS  512
#define D_OUT     512
#define D_IN      1024
#define NOISE_DIM 64
#define M_SAMP    16
#define P_PITCH   (D_OUT + 8)   // pad LDS rows: bank = (8*m + d) % 64, avoids 16-way conflicts

// ---------------------------------------------------------------------------
// K0: gid[i] = argmax_g agg[g, i]  (agg is one-hot [G, n], row-major)
// ---------------------------------------------------------------------------
__global__ void gid_kernel(const float* __restrict__ agg, int* __restrict__ gid) {
    int i = blockIdx.x * blockDim.x + threadIdx.x;
    if (i >= N_ROWS) return;
    int g = 0;
    for (int gg = 0; gg < G_GROUPS; ++gg) {
        float v = agg[(size_t)gg * N_ROWS + i];   // coalesced across i
        if (v > 0.5f) g = gg;
    }
    gid[i] = g;
}

// ---------------------------------------------------------------------------
// K1: ordered member lists per group (deterministic aggregation order)
// ---------------------------------------------------------------------------
__global__ void group_lists_kernel(const int* __restrict__ gid,
                                   int* __restrict__ counts,
                                   int* __restrict__ offsets,
                                   int* __restrict__ mlist) {
    int g = threadIdx.x;                 // 512 threads, one per group
    int cnt = 0;
    for (int i = 0; i < N_ROWS; ++i) cnt += (gid[i] == g) ? 1 : 0;
    __shared__ int sc[G_GROUPS];
    __shared__ int soff[G_GROUPS];
    sc[g] = cnt;
    counts[g] = cnt;
    __syncthreads();
    if (g == 0) {
        int acc = 0;
        for (int k = 0; k < G_GROUPS; ++k) { soff[k] = acc; acc += sc[k]; }
    }
    __syncthreads();
    int off = soff[g];
    offsets[g] = off;
    for (int i = 0; i < N_ROWS; ++i)
        if (gid[i] == g) mlist[off++] = i;   // ascending i per group
}

// ---------------------------------------------------------------------------
// K2: Y[4096,512] = x[4096,1024] @ W[:1024,:512]   (WMMA f32 16x16x4)
// grid (32 tileN, 32 rowGroups of 128), block (32,8): wave w owns 16 M-rows,
// all 8 waves share the LDS-staged B slab for their common tileN.
// ---------------------------------------------------------------------------
#define KCHUNK 128
__global__ __launch_bounds__(256) void ygemm_kernel(const float* __restrict__ x,
                                                    const float* __restrict__ W,
                                                    float* __restrict__ Y) {
    __shared__ float Bs[KCHUNK][16];
    const int lane  = threadIdx.x;
    const int wave  = threadIdx.y;
    const int tid   = wave * 32 + lane;
    const int tileN = blockIdx.x;
    const int rowBase = blockIdx.y * 128 + wave * 16;
    const int row = lane & 15, kh = lane >> 4, col = lane & 15;

    const float* Arow = x + (size_t)(rowBase + row) * D_IN;
    v8f c = {0.f, 0.f, 0.f, 0.f, 0.f, 0.f, 0.f, 0.f};

    for (int k0 = 0; k0 < D_IN; k0 += KCHUNK) {
        __syncthreads();
        for (int idx = tid; idx < KCHUNK * 16; idx += 256) {
            int r = idx >> 4, cc = idx & 15;
            Bs[r][cc] = W[(size_t)(k0 + r) * D_OUT + tileN * 16 + cc];
        }
        __syncthreads();
        #pragma unroll 4
        for (int k = 0; k < KCHUNK; k += 4) {
            int ka = k + 2 * kh;
            v2f a; a.x = Arow[k0 + ka];  a.y = Arow[k0 + ka + 1];
            v2f b; b.x = Bs[ka][col];    b.y = Bs[ka + 1][col];
            c = __builtin_amdgcn_wmma_f32_16x16x4_f32(false, a, false, b,
                                                      (short)0, c, false, false);
        }
    }
    #pragma unroll
    for (int j = 0; j < 8; ++j)
        Y[(size_t)(rowBase + j + 8 * kh) * D_OUT + tileN * 16 + col] = c[j];
}

__device__ __forceinline__ float softplus_f(float v) {
    // jax.nn.softplus: max(v,0) + log1p(exp(-|v|))  (stable)
    return fmaxf(v, 0.0f) + log1pf(expf(-fabsf(v)));
}

// ---------------------------------------------------------------------------
// K3: one block per group. For each member i: WMMA noise[16x64] @ Wn[64x512],
// add Y[i]+b, softplus, accumulate (fixed order) -> P[16,512] in LDS; then
// confinement norms + Gram + clamped pairwise distances -> perGroup[g].
// ---------------------------------------------------------------------------
__global__ __launch_bounds__(256) void group_energy_kernel(
        const float* __restrict__ target, const float* __restrict__ noise,
        const float* __restrict__ W,      const float* __restrict__ bias,
        const float* __restrict__ Y,      const int* __restrict__ counts,
        const int* __restrict__ offsets,  const int* __restrict__ mlist,
        float* __restrict__ perGroup) {
    const int g    = blockIdx.x;
    const int tid  = threadIdx.x;
    const int lane = tid & 31;
    const int wave = tid >> 5;
    const int row = lane & 15, kh = lane >> 4, col = lane & 15;
    const float* Wn = W + (size_t)D_IN * D_OUT;     // [64, 512]
    const v8f vzero = {0.f, 0.f, 0.f, 0.f, 0.f, 0.f, 0.f, 0.f};

    __shared__ float P[M_SAMP * P_PITCH];           // ~33 KB, padded rows
    __shared__ float conf[M_SAMP];
    __shared__ float gram[M_SAMP * M_SAMP];
    __shared__ float red[256];

    const int nm  = counts[g];
    const int off = offsets[g];

    // each wave owns 4 N-tiles of 16 columns; hoist B fragments per tile.
    for (int t = 0; t < 4; ++t) {
        const int cg = (wave * 4 + t) * 16 + col;
        v2f bf[NOISE_DIM / 4];
        #pragma unroll
        for (int ks = 0; ks < NOISE_DIM / 4; ++ks) {
            int ka = ks * 4 + 2 * kh;
            bf[ks].x = Wn[(size_t)ka * D_OUT + cg];
            bf[ks].y = Wn[(size_t)(ka + 1) * D_OUT + cg];
        }
        const float bcol = bias[cg];
        v8f acc = vzero;
        for (int mi = 0; mi < nm; ++mi) {
            const int i = mlist[off + mi];
            const float* Arow = noise + (size_t)(i * M_SAMP + row) * NOISE_DIM;
            v8f c = vzero;
            #pragma unroll
            for (int ks = 0; ks < NOISE_DIM / 4; ++ks) {
                int ka = ks * 4 + 2 * kh;
                v2f a; a.x = Arow[ka]; a.y = Arow[ka + 1];
                c = __builtin_amdgcn_wmma_f32_16x16x4_f32(false, a, false, bf[ks],
                                                          (short)0, c, false, false);
            }
            const float base = Y[(size_t)i * D_OUT + cg] + bcol;
            #pragma unroll
            for (int j = 0; j < 8; ++j)
                acc[j] += softplus_f(c[j] + base);
        }
        #pragma unroll
        for (int j = 0; j < 8; ++j)
            P[(j + 8 * kh) * P_PITCH + cg] = acc[j];
    }
    __syncthreads();

    // confinement: wave w handles m = 2w, 2w+1
    for (int m = wave * 2; m < wave * 2 + 2; ++m) {
        float s = 0.f;
        for (int d = lane; d < D_OUT; d += 32) {
            float diff = P[m * P_PITCH + d] - target[(size_t)g * D_OUT + d];
            s += diff * diff;
        }
        #pragma unroll
        for (int o = 16; o > 0; o >>= 1) s += __shfl_xor(s, o, 32);
        if (lane == 0) conf[m] = sqrtf(s);
    }

    // Gram: thread (m,n) computes <P_m, P_n>
    {
        const int mm = tid >> 4, nn = tid & 15;
        float s = 0.f;
        for (int d = 0; d < D_OUT; ++d)
            s += P[mm * P_PITCH + d] * P[nn * P_PITCH + d];
        gram[mm * 16 + nn] = s;
    }
    __syncthreads();

    // pairwise distances with clamp, off-diagonal sum
    {
        const int mm = tid >> 4, nn = tid & 15;
        float dsum = 0.f;
        if (mm != nn) {
            float sqd = gram[mm * 16 + mm] + gram[nn * 16 + nn]
                      - 2.f * gram[mm * 16 + nn];
            dsum = sqrtf(fmaxf(sqd, 1e-6f));
        }
        red[tid] = dsum;
    }
    __syncthreads();
    for (int sN = 128; sN > 0; sN >>= 1) {
        if (tid < sN) red[tid] += red[tid + sN];
        __syncthreads();
    }
    if (tid == 0) {
        float cs = 0.f;
        #pragma unroll
        for (int m = 0; m < M_SAMP; ++m) cs += conf[m];
        float term_conf = cs * (1.0f / M_SAMP);
        float mean_pd   = red[0] * (1.0f / (M_SAMP * (M_SAMP - 1)));
        perGroup[g] = term_conf - 0.5f * mean_pd;   // LAMBDA = 1
    }
}

// ---------------------------------------------------------------------------
// K4: deterministic mean over 512 groups
// ---------------------------------------------------------------------------
__global__ void final_reduce_kernel(const float* __restrict__ perGroup,
                                    float* __restrict__ out) {
    __shared__ float red[256];
    int tid = threadIdx.x;
    red[tid] = perGroup[tid] + perGroup[tid + 256];
    __syncthreads();
    for (int sN = 128; sN > 0; sN >>= 1) {
        if (tid < sN) red[tid] += red[tid + sN];
        __syncthreads();
    }
    if (tid == 0) out[0] = red[0] * (1.0f / G_GROUPS);
}

// ---------------------------------------------------------------------------
extern "C" void kernel_launch(void* const* d_in, const int* in_sizes, int n_in,
                              void* d_out, int out_size, void* d_ws, size_t ws_size,
                              hipStream_t stream) {
    (void)in_sizes; (void)n_in; (void)out_size; (void)ws_size;
    const float* target = (const float*)d_in[0];   // [512, 512]
    const float* x      = (const float*)d_in[1];   // [4096, 1024]
    const float* agg    = (const float*)d_in[2];   // [512, 4096] one-hot
    const float* noise  = (const float*)d_in[3];   // [65536, 64]
    const float* W      = (const float*)d_in[4];   // [1088, 512]
    const float* bias   = (const float*)d_in[5];   // [512]
    float* out = (float*)d_out;

    char* ws = (char*)d_ws;
    size_t o = 0;
    auto take = [&](size_t bytes) { char* p = ws + o; o += (bytes + 255) & ~size_t(255); return p; };
    int*   gid      = (int*)  take(N_ROWS * sizeof(int));
    int*   counts   = (int*)  take(G_GROUPS * sizeof(int));
    int*   offsets  = (int*)  take(G_GROUPS * sizeof(int));
    int*   mlist    = (int*)  take(N_ROWS * sizeof(int));
    float* Y        = (float*)take((size_t)N_ROWS * D_OUT * sizeof(float));
    float* perGroup = (float*)take(G_GROUPS * sizeof(float));

    gid_kernel<<<N_ROWS / 256, 256, 0, stream>>>(agg, gid);
    group_lists_kernel<<<1, G_GROUPS, 0, stream>>>(gid, counts, offsets, mlist);
    ygemm_kernel<<<dim3(D_OUT / 16, N_ROWS / 128), dim3(32, 8), 0, stream>>>(x, W, Y);
    group_energy_kernel<<<G_GROUPS, 256, 0, stream>>>(target, noise, W, bias, Y,
                                                      counts, offsets, mlist, perGroup);
    final_reduce_kernel<<<1, 256, 0, stream>>>(perGroup, out);
}